// EGAT_with_DIFFPool_25245817766263
// MI455X (gfx1250) — compile-verified
//
#include <hip/hip_runtime.h>
#include <math.h>

// ---- problem constants (from reference) ----
#define NNODES 3072
#define NE     98304
#define EPL    101376           // NE + NNODES (with self loops)
#define FIN    128
#define H1C    4
#define D1C    8
#define EC1    3
#define EC2C   12               // H1*EC
#define D2C    8
#define NEGS   0.2f

typedef float v2f __attribute__((ext_vector_type(2)));
typedef float v8f __attribute__((ext_vector_type(8)));

__device__ __forceinline__ v8f wmma4f32(v2f a, v2f b, v8f c) {
  // D = A(16x4) * B(4x16) + C(16x16), fp32 WMMA (CDNA5)
  return __builtin_amdgcn_wmma_f32_16x16x4_f32(false, a, false, b, (short)0, c,
                                               false, false);
}

// monotonic float<->uint encoding for atomic segment-max
__device__ __forceinline__ unsigned encf(float f) {
  int i = __float_as_int(f);
  return (i >= 0) ? ((unsigned)i | 0x80000000u) : (unsigned)(~i);
}
__device__ __forceinline__ float decf(unsigned u) {
  int i = (u & 0x80000000u) ? (int)(u ^ 0x80000000u) : (int)(~u);
  return __int_as_float(i);
}

__device__ __forceinline__ float eluf(float v) {
  return v > 0.f ? v : (expf(v) - 1.f);
}

// ============================================================================
// Generic f32 WMMA GEMM: C[M,N] = A * B, A indexed A[m*as_r + k*as_c],
// B row-major ldb, C row-major ldc. Optional batching via blockIdx.y.
// Requirements: M % 16 == 0, K % 16 == 0. N guarded (column-clamped loads:
// out-of-range B columns only affect out-of-range D columns, never stored).
// ============================================================================
__global__ void wmma_gemm_f32(const float* __restrict__ A,
                              const float* __restrict__ B,
                              float* __restrict__ C,
                              int M, int N, int K,
                              int as_r, int as_c, int ldb, int ldc,
                              long long sA, long long sB, long long sC) {
  const int ntiles = (N + 15) >> 4;
  const int mtiles = M >> 4;
  const int wave = blockIdx.x * (blockDim.x >> 5) + (threadIdx.x >> 5);
  if (wave >= mtiles * ntiles) return;
  const int tm = wave / ntiles, tn = wave - tm * ntiles;
  const int lane = threadIdx.x & 31, lo = lane & 15, hi = lane >> 4;

  A += (long long)blockIdx.y * sA;
  B += (long long)blockIdx.y * sB;
  C += (long long)blockIdx.y * sC;

  const int m = tm * 16 + lo;                 // A row held by this lane
  const int n = tn * 16 + lo;                 // B/C column held by this lane
  const bool nok = (n < N);
  const int ncl = nok ? n : (N - 1);          // clamp: avoids OOB, no exec churn

  const float* ap = A + (long long)m * as_r + (long long)(2 * hi) * as_c;
  const float* bp = B + (long long)(2 * hi) * ldb + ncl;
  const long long astep = (long long)4 * as_c;
  const long long bstep = (long long)4 * ldb;

  v8f acc = {};
  for (int k0 = 0; k0 < K; k0 += 16) {
    v2f a[4], b[4];
#pragma unroll
    for (int u = 0; u < 4; ++u) {
      a[u].x = ap[u * astep];
      a[u].y = ap[u * astep + as_c];
      b[u].x = bp[u * bstep];
      b[u].y = bp[u * bstep + ldb];
    }
#pragma unroll
    for (int u = 0; u < 4; ++u) acc = wmma4f32(a[u], b[u], acc);
    ap += 4 * astep;
    bp += 4 * bstep;
  }
  if (nok) {
    const int rbase = tm * 16 + 8 * hi;
#pragma unroll
    for (int r = 0; r < 8; ++r)
      C[(long long)(rbase + r) * ldc + n] = acc[r];
  }
}

// ============================================================================
// Fused DiffPool-1 streaming kernel over adj [12,3072,3072]:
//   T[c,i,m] = sum_n s[n,i] * adj[c,n,m]   (WMMA, A = s^T)
//   *sumsq  += sum adj^2                    (for link loss, algebraic form)
// One non-temporal pass over 452 MB -> bandwidth bound. grid=(24,12),
// block=256 (8 waves), each wave owns 16 output columns.
// ============================================================================
__global__ void adj_pool(const float* __restrict__ adj,
                         const float* __restrict__ s,
                         float* __restrict__ T,
                         float* __restrict__ sumsq) {
  const int c = blockIdx.y;
  const int wib = threadIdx.x >> 5;
  const int m0 = blockIdx.x * 128 + wib * 16;
  const int lane = threadIdx.x & 31, lo = lane & 15, hi = lane >> 4;

  // lane's adj pointer: row (2*hi), column (m0+lo); rows advance by NNODES
  const float* bp = adj + (size_t)c * NNODES * NNODES +
                    (size_t)(2 * hi) * NNODES + (m0 + lo);
  const float* ap = s + (2 * hi) * 16 + lo;   // s[n*16 + lo] = A[i=lo, k=n]

  v8f acc = {};
  float ss = 0.f;
  const size_t rstep = (size_t)4 * NNODES;    // 4 rows per sub-step
  for (int n0 = 0; n0 < NNODES; n0 += 16) {
    v2f a[4], b[4];
#pragma unroll
    for (int u = 0; u < 4; ++u) {
      a[u].x = ap[u * 64];
      a[u].y = ap[u * 64 + 16];
      b[u].x = __builtin_nontemporal_load(&bp[u * rstep]);
      b[u].y = __builtin_nontemporal_load(&bp[u * rstep + NNODES]);
    }
#pragma unroll
    for (int u = 0; u < 4; ++u) {
      acc = wmma4f32(a[u], b[u], acc);
      ss = fmaf(b[u].x, b[u].x, ss);
      ss = fmaf(b[u].y, b[u].y, ss);
    }
    ap += 256;                                 // 16 rows of s
    bp += 4 * rstep;                           // 16 rows of adj
  }
#pragma unroll
  for (int r = 0; r < 8; ++r)
    T[((size_t)c * 16 + r + 8 * hi) * NNODES + (m0 + lo)] = acc[r];

  for (int off = 16; off; off >>= 1) ss += __shfl_xor(ss, off, 32);
  if (lane == 0) atomicAdd(sumsq, ss);
}

// ============================================================================
// Per-node attention projections: ps[n,h]=h.a[:D], pd[n,h]=h.a[D:2D]
// ============================================================================
__global__ void node_proj(const float* __restrict__ hlin,
                          const float* __restrict__ av,
                          float* __restrict__ ps, float* __restrict__ pd,
                          int H, int D) {
  const int n = blockIdx.x * blockDim.x + threadIdx.x;
  if (n >= NNODES) return;
  for (int h = 0; h < H; ++h) {
    float s1 = 0.f, s2 = 0.f;
    for (int d = 0; d < D; ++d) {
      const float v = hlin[(size_t)n * H * D + h * D + d];
      s1 = fmaf(v, av[h * 2 * D + d], s1);
      s2 = fmaf(v, av[h * 2 * D + D + d], s2);
    }
    ps[n * H + h] = s1;
    pd[n * H + h] = s2;
  }
}

// ============================================================================
// Edge passes (self-loops are implicit edges e >= NE with r=c=e-NE)
// ============================================================================
__global__ void edge_alpha(const int* __restrict__ ei,
                           const float* __restrict__ ps,
                           const float* __restrict__ pd,
                           float* __restrict__ ab,
                           unsigned* __restrict__ amax, int H) {
  const int e = blockIdx.x * blockDim.x + threadIdx.x;
  if (e >= EPL) return;
  int r, c;
  if (e < NE) { r = ei[e]; c = ei[NE + e]; } else { r = c = e - NE; }
  for (int h = 0; h < H; ++h) {
    float a = ps[r * H + h] + pd[c * H + h];
    a = (a > 0.f) ? a : NEGS * a;            // leaky relu
    ab[(size_t)e * H + h] = a;
    atomicMax(&amax[r * H + h], encf(a));
  }
}

__global__ void edge_exp(const int* __restrict__ ei,
                         float* __restrict__ ab,
                         const unsigned* __restrict__ amax,
                         float* __restrict__ denom, int H) {
  const int e = blockIdx.x * blockDim.x + threadIdx.x;
  if (e >= EPL) return;
  const int r = (e < NE) ? ei[e] : (e - NE);
  for (int h = 0; h < H; ++h) {
    const float ex = expf(ab[(size_t)e * H + h] - decf(amax[r * H + h]));
    ab[(size_t)e * H + h] = ex;
    atomicAdd(&denom[r * H + h], ex);
  }
}

// one thread per (edge, head, edge-channel); scatters D=8 message values
__global__ void edge_scatter(const int* __restrict__ ei,
                             const float* __restrict__ ab,
                             const float* __restrict__ denom,
                             const float* __restrict__ ea,
                             const float* __restrict__ hsrc,
                             float* __restrict__ out,
                             float* __restrict__ eout,
                             int H, int CE, int D, int hd_ld, int ea_full) {
  const int HC = H * CE;
  const int t = blockIdx.x * blockDim.x + threadIdx.x;
  if (t >= EPL * HC) return;
  const int e = t / HC, hc = t - e * HC;
  const int h = hc / CE, cc = hc - h * CE;
  int r, c;
  if (e < NE) { r = ei[e]; c = ei[NE + e]; } else { r = c = e - NE; }
  const float al = ab[(size_t)e * H + h] / (denom[r * H + h] + 1e-16f);
  float eav;
  if (ea_full) eav = ea[(size_t)e * CE + cc];
  else         eav = (e < NE) ? ea[(size_t)e * CE + cc] : 1.0f;
  const float attv = al * eav;
  if (eout) eout[(size_t)e * HC + hc] = attv;
  const float* hv = hsrc + (size_t)c * hd_ld + h * D;
  float* op = out + (size_t)r * (HC * D) + hc * D;
  for (int d = 0; d < D; ++d) atomicAdd(&op[d], attv * hv[d]);
}

__global__ void elu_k(float* __restrict__ x, int n) {
  const int i = blockIdx.x * blockDim.x + threadIdx.x;
  if (i < n) x[i] = eluf(x[i]);
}

// softmax over rows of s1 [3072,16] + entropy accumulation
__global__ void softmax_s(const float* __restrict__ s1,
                          float* __restrict__ ss, float* __restrict__ entacc) {
  const int n = blockIdx.x * blockDim.x + threadIdx.x;
  if (n >= NNODES) return;
  float v[16], mx = -3.0e38f;
  for (int i = 0; i < 16; ++i) { v[i] = s1[n * 16 + i]; mx = fmaxf(mx, v[i]); }
  float sum = 0.f;
  for (int i = 0; i < 16; ++i) { v[i] = expf(v[i] - mx); sum += v[i]; }
  const float inv = 1.f / sum;
  float ent = 0.f;
  for (int i = 0; i < 16; ++i) {
    const float p = v[i] * inv;
    ss[n * 16 + i] = p;
    ent = fmaf(p, logf(p + 1e-15f), ent);
  }
  atomicAdd(entacc, ent);
}

// ============================================================================
// Pool-2 (+ FC head + both regularizers). softmax(s2 [16,1]) == ones, so
// x2 = column-sum of x1; reg2 = sqrt(sum (adj1-1)^2)/3072 - log1p(1e-15).
// reg1 = sqrt(sum adj^2 - 2 tr(sum_c adjn_c) + 12 ||G||_F^2)/adj.size + ent1.
// Single 128-thread block.
// ============================================================================
__global__ void head_kernel(const float* __restrict__ x1,
                            const float* __restrict__ adjn,
                            const float* __restrict__ G,
                            const float* __restrict__ scal,
                            const float* __restrict__ fc1w, const float* __restrict__ fc1b,
                            const float* __restrict__ fc2w, const float* __restrict__ fc2b,
                            const float* __restrict__ fc3w, const float* __restrict__ fc3b,
                            const float* __restrict__ fc4w, const float* __restrict__ fc4b,
                            float* __restrict__ out) {
  __shared__ float t0[96], t1[128], t2[32], t3[16], lg[2], red[128];
  const int tid = threadIdx.x;

  if (tid < 96) {
    float s = 0.f;
    for (int i = 0; i < 16; ++i) s += x1[i * 96 + tid];
    t0[tid] = s;
  }
  __syncthreads();
  {
    float s = fc1b[tid];
    for (int f = 0; f < 96; ++f) s = fmaf(t0[f], fc1w[f * 128 + tid], s);
    t1[tid] = eluf(s);
  }
  __syncthreads();
  if (tid < 32) {
    float s = fc2b[tid];
    for (int f = 0; f < 128; ++f) s = fmaf(t1[f], fc2w[f * 32 + tid], s);
    t2[tid] = eluf(s);
  }
  __syncthreads();
  if (tid < 16) {
    float s = fc3b[tid];
    for (int f = 0; f < 32; ++f) s = fmaf(t2[f], fc3w[f * 16 + tid], s);
    t3[tid] = eluf(s);
  }
  __syncthreads();
  if (tid < 2) {
    float s = fc4b[tid];
    for (int f = 0; f < 16; ++f) s = fmaf(t3[f], fc4w[f * 2 + tid], s);
    lg[tid] = s;
  }
  __syncthreads();

  // reg2: sum over 12*16*16 of (adjn - 1)^2
  float ssq = 0.f;
  for (int i = tid; i < 12 * 256; i += 128) {
    const float d = adjn[i] - 1.f;
    ssq = fmaf(d, d, ssq);
  }
  red[tid] = ssq;
  __syncthreads();
  for (int off = 64; off; off >>= 1) {
    if (tid < off) red[tid] += red[tid + off];
    __syncthreads();
  }
  float reg2 = 0.f;
  if (tid == 0) reg2 = sqrtf(red[0]) / 3072.f - log1pf(1e-15f);
  __syncthreads();

  // ||G||_F^2 (G = s^T s, 16x16)
  float g = 0.f;
  for (int i = tid; i < 256; i += 128) { const float v = G[i]; g = fmaf(v, v, g); }
  red[tid] = g;
  __syncthreads();
  for (int off = 64; off; off >>= 1) {
    if (tid < off) red[tid] += red[tid + off];
    __syncthreads();
  }

  if (tid == 0) {
    const float gsq = red[0];
    float tr = 0.f;
    for (int c = 0; c < 12; ++c)
      for (int i = 0; i < 16; ++i) tr += adjn[c * 256 + i * 16 + i];
    float term = scal[0] - 2.f * tr + 12.f * gsq;
    term = fmaxf(term, 0.f);
    const float link = sqrtf(term) / (12.f * 3072.f * 3072.f);
    const float ent1 = -(scal[1] / 3072.f);

    const float m = fmaxf(lg[0], lg[1]);
    const float lse = m + logf(expf(lg[0] - m) + expf(lg[1] - m));
    out[0] = lg[0] - lse;
    out[1] = lg[1] - lse;
    out[2] = link + ent1;
    out[3] = reg2;
  }
}

// ============================================================================
// host-side launcher
// ============================================================================
static inline void launch_gemm(const float* A, const float* B, float* C,
                               int M, int N, int K, int as_r, int as_c,
                               int ldb, int ldc, int batch,
                               long long sA, long long sB, long long sC,
                               hipStream_t st) {
  const int tiles = (M / 16) * ((N + 15) / 16);
  const int wpb = 4;
  dim3 grid((tiles + wpb - 1) / wpb, batch, 1);
  wmma_gemm_f32<<<grid, wpb * 32, 0, st>>>(A, B, C, M, N, K, as_r, as_c, ldb,
                                           ldc, sA, sB, sC);
}

extern "C" void kernel_launch(void* const* d_in, const int* in_sizes, int n_in,
                              void* d_out, int out_size, void* d_ws,
                              size_t ws_size, hipStream_t stream) {
  (void)in_sizes; (void)n_in; (void)out_size; (void)ws_size;
  const float* x    = (const float*)d_in[0];
  const int*   ei   = (const int*)d_in[1];
  const float* ea   = (const float*)d_in[2];
  const float* adj  = (const float*)d_in[3];
  const float* w1   = (const float*)d_in[4];
  const float* a1   = (const float*)d_in[5];
  const float* w2   = (const float*)d_in[6];
  const float* a2   = (const float*)d_in[7];
  const float* s1   = (const float*)d_in[8];
  /* d_in[9] = s2: softmax over a size-1 axis == ones -> unused */
  const float* fc1w = (const float*)d_in[10];
  const float* fc1b = (const float*)d_in[11];
  const float* fc2w = (const float*)d_in[12];
  const float* fc2b = (const float*)d_in[13];
  const float* fc3w = (const float*)d_in[14];
  const float* fc3b = (const float*)d_in[15];
  const float* fc4w = (const float*)d_in[16];
  const float* fc4b = (const float*)d_in[17];
  float* out = (float*)d_out;
  float* ws = (float*)d_ws;

  // ---- workspace layout (floats, 256B aligned). Zeroed region comes first.
  size_t off = 0;
  auto alloc = [&](size_t n) { size_t o = off; off += (n + 63) & ~(size_t)63; return o; };
  const size_t o_amax1  = alloc((size_t)NNODES * H1C);   // unsigned
  const size_t o_den1   = alloc((size_t)NNODES * H1C);
  const size_t o_amax2  = alloc((size_t)NNODES);         // unsigned
  const size_t o_den2   = alloc((size_t)NNODES);
  const size_t o_hout1  = alloc((size_t)NNODES * 96);
  const size_t o_hout2  = alloc((size_t)NNODES * 96);
  const size_t o_scal   = alloc(64);
  const size_t zero_floats = off;                        // memset range
  const size_t o_h1lin  = alloc((size_t)NNODES * 32);
  const size_t o_p1s    = alloc((size_t)NNODES * H1C);
  const size_t o_p1d    = alloc((size_t)NNODES * H1C);
  const size_t o_alpha1 = alloc((size_t)EPL * H1C);
  const size_t o_e1     = alloc((size_t)EPL * EC2C);
  const size_t o_h2lin  = alloc((size_t)NNODES * 8);
  const size_t o_p2s    = alloc((size_t)NNODES);
  const size_t o_p2d    = alloc((size_t)NNODES);
  const size_t o_alpha2 = alloc((size_t)EPL);
  const size_t o_ssoft  = alloc((size_t)NNODES * 16);
  const size_t o_T      = alloc((size_t)12 * 16 * NNODES);
  const size_t o_adjn   = alloc((size_t)12 * 256);
  const size_t o_x1     = alloc((size_t)16 * 96);
  const size_t o_G      = alloc((size_t)256);

  hipMemsetAsync(ws, 0, zero_floats * sizeof(float), stream);

  const int EB = (EPL + 255) / 256;
  const int NB = (NNODES + 255) / 256;

  // ---- EGAT layer 1 ----
  launch_gemm(x, w1, ws + o_h1lin, NNODES, 32, FIN, FIN, 1, 32, 32, 1, 0, 0, 0, stream);
  node_proj<<<NB, 256, 0, stream>>>(ws + o_h1lin, a1, ws + o_p1s, ws + o_p1d, H1C, D1C);
  edge_alpha<<<EB, 256, 0, stream>>>(ei, ws + o_p1s, ws + o_p1d, ws + o_alpha1,
                                     (unsigned*)(ws + o_amax1), H1C);
  edge_exp<<<EB, 256, 0, stream>>>(ei, ws + o_alpha1, (const unsigned*)(ws + o_amax1),
                                   ws + o_den1, H1C);
  {
    const int cnt = EPL * H1C * EC1;
    edge_scatter<<<(cnt + 255) / 256, 256, 0, stream>>>(
        ei, ws + o_alpha1, ws + o_den1, ea, ws + o_h1lin, ws + o_hout1,
        ws + o_e1, H1C, EC1, D1C, 32, /*ea_full=*/0);
  }
  elu_k<<<(NNODES * 96 + 255) / 256, 256, 0, stream>>>(ws + o_hout1, NNODES * 96);

  // ---- EGAT layer 2 ----
  launch_gemm(ws + o_hout1, w2, ws + o_h2lin, NNODES, 8, 96, 96, 1, 8, 8, 1, 0, 0, 0, stream);
  node_proj<<<NB, 256, 0, stream>>>(ws + o_h2lin, a2, ws + o_p2s, ws + o_p2d, 1, D2C);
  edge_alpha<<<EB, 256, 0, stream>>>(ei, ws + o_p2s, ws + o_p2d, ws + o_alpha2,
                                     (unsigned*)(ws + o_amax2), 1);
  edge_exp<<<EB, 256, 0, stream>>>(ei, ws + o_alpha2, (const unsigned*)(ws + o_amax2),
                                   ws + o_den2, 1);
  {
    const int cnt = EPL * 1 * EC2C;
    edge_scatter<<<(cnt + 255) / 256, 256, 0, stream>>>(
        ei, ws + o_alpha2, ws + o_den2, ws + o_e1, ws + o_h2lin, ws + o_hout2,
        (float*)nullptr, 1, EC2C, D2C, 8, /*ea_full=*/1);
  }
  elu_k<<<(NNODES * 96 + 255) / 256, 256, 0, stream>>>(ws + o_hout2, NNODES * 96);

  // ---- DiffPool 1 ----
  softmax_s<<<NB, 256, 0, stream>>>(s1, ws + o_ssoft, ws + o_scal + 1);
  {
    dim3 grid(NNODES / 128, 12, 1);
    adj_pool<<<grid, 256, 0, stream>>>(adj, ws + o_ssoft, ws + o_T, ws + o_scal + 0);
  }
  // adj_new[c] = T[c] @ s  (batched, 12 channels)
  launch_gemm(ws + o_T, ws + o_ssoft, ws + o_adjn, 16, 16, NNODES, NNODES, 1,
              16, 16, 12, (long long)16 * NNODES, 0, 256, stream);
  // x1 = s^T @ h2
  launch_gemm(ws + o_ssoft, ws + o_hout2, ws + o_x1, 16, 96, NNODES, 1, 16,
              96, 96, 1, 0, 0, 0, stream);
  // G = s^T @ s
  launch_gemm(ws + o_ssoft, ws + o_ssoft, ws + o_G, 16, 16, NNODES, 1, 16,
              16, 16, 1, 0, 0, 0, stream);

  // ---- Pool 2 + FC head + regularizers ----
  head_kernel<<<1, 128, 0, stream>>>(ws + o_x1, ws + o_adjn, ws + o_G,
                                     ws + o_scal, fc1w, fc1b, fc2w, fc2b,
                                     fc3w, fc3b, fc4w, fc4b, out);
}